// ISACNet_35313221108057
// MI455X (gfx1250) — compile-verified
//
#include <hip/hip_runtime.h>
#include <hip/hip_bf16.h>
#include <math.h>

typedef float v2f __attribute__((ext_vector_type(2)));
typedef float v8f __attribute__((ext_vector_type(8)));
typedef int   v4i __attribute__((ext_vector_type(4)));
typedef __attribute__((address_space(1))) v4i* gv4i_p;   // global
typedef __attribute__((address_space(3))) v4i* lv4i_p;   // LDS

#if defined(__has_builtin)
#if __has_builtin(__builtin_amdgcn_global_load_async_to_lds_b128) && \
    __has_builtin(__builtin_amdgcn_s_wait_asynccnt)
#define USE_ASYNC 1
#endif
#endif
#ifndef USE_ASYNC
#define USE_ASYNC 0
#endif

#if USE_ASYNC
#define ASYNC_WAIT() __builtin_amdgcn_s_wait_asynccnt(0)
#else
#define ASYNC_WAIT()
#endif

#define BATCH   8192
#define DIN     9216
#define H1N     64
#define H2N     32
#define KC      128            // K chunk for GEMM1
#define NCHUNK  (DIN / KC)     // 72
#define ROWS    16             // batch rows per block
#define XS_PITCH 132           // 128 + 4 pad (bank spread)
#define H1_PITCH 68
#define H2_PITCH 36
#define HH_PITCH 724           // 720 + 4 pad
// head buffer section offsets within a row of hh
#define OFF_P   0              // 128
#define OFF_PH  128            // 512
#define OFF_Z   640            // 16
#define OFF_B   656            // 64
// output flat offsets (floats)
#define W_OFF   0ull                         // 8192*512 complex = 8388608 floats
#define Z_OFF   8388608ull                   // 16*4*4 complex = 512 floats
#define B_OFF   8389120ull                   // 8192*64 floats

__device__ __forceinline__ float sigm(float v) { return 1.0f / (1.0f + __expf(-v)); }

// Stage one 16x128 f32 chunk of x into LDS. xg points at x[row0*DIN + kc].
// Async path: global -> LDS directly (ASYNCcnt); fallback: float4 via VGPRs.
__device__ __forceinline__ void stage_chunk(const float* __restrict__ xg,
                                            float* dst, int tid) {
    #pragma unroll
    for (int j = 0; j < 4; ++j) {
        int i  = tid + j * 128;         // 0..511 float4 slots
        int r  = i >> 5;                // row (32 float4 per row)
        int c4 = (i & 31) << 2;         // col (floats)
#if USE_ASYNC
        __builtin_amdgcn_global_load_async_to_lds_b128(
            (gv4i_p)(xg + (size_t)r * DIN + c4),
            (lv4i_p)(dst + r * XS_PITCH + c4),
            /*offset=*/0, /*cpol=*/0);
#else
        float4 v = *(const float4*)(xg + (size_t)r * DIN + c4);
        *(float4*)(dst + r * XS_PITCH + c4) = v;
#endif
    }
}

__global__ __launch_bounds__(128)
void isacnet_fused(const float* __restrict__ x,
                   const float* __restrict__ W1, const float* __restrict__ b1,
                   const float* __restrict__ W2, const float* __restrict__ b2,
                   const float* __restrict__ Wp, const float* __restrict__ bp,
                   const float* __restrict__ Wph, const float* __restrict__ bph,
                   const float* __restrict__ Wz, const float* __restrict__ bz,
                   const float* __restrict__ Wb, const float* __restrict__ bb,
                   float* __restrict__ out) {
    __shared__ __align__(16) float xs[2][ROWS * XS_PITCH]; // double-buffered x chunk
    __shared__ __align__(16) float h1s[ROWS * H1_PITCH];   // h1 tile 16x64
    __shared__ __align__(16) float h2s[ROWS * H2_PITCH];   // h2 tile 16x32
    __shared__ __align__(16) float hh[ROWS * HH_PITCH];    // head preacts 16x720
    __shared__ float partial[ROWS * 8];
    __shared__ float rowsum[ROWS];

    const int tid  = threadIdx.x;
    const int wave = tid >> 5;
    const int lane = tid & 31;
    const int ml   = lane & 15;         // M (or N) index within tile
    const int krow = (lane >> 4) << 1;  // 0 or 2: K sub-row pair
    const int rbase = (lane >> 4) << 3; // C-fragment M base (0 or 8)
    const int row0 = blockIdx.x * ROWS; // first batch row of this block
    const float* xblk = x + (size_t)row0 * DIN;

    // ---------------- GEMM1: h1 = leakyrelu(x @ W1 + b1), tile 16x64 ----------
    // Double-buffered pipeline: stage chunk c+1 while WMMAs consume chunk c.
    const int n0 = wave * 16;           // this wave's W1 column tile
    v8f acc = {};

    stage_chunk(xblk, xs[0], tid);      // preload chunk 0
    ASYNC_WAIT();
    __syncthreads();

    for (int c = 0; c < NCHUNK; ++c) {
        if (c + 1 < NCHUNK)
            stage_chunk(xblk + (size_t)(c + 1) * KC, xs[(c + 1) & 1], tid);

        const float* cur = xs[c & 1];
        const int kc = c * KC;
        #pragma unroll 8
        for (int k4 = 0; k4 < KC; k4 += 4) {
            float2 af = *(const float2*)(&cur[ml * XS_PITCH + k4 + krow]);
            v2f a; a.x = af.x; a.y = af.y;
            int kk = kc + k4 + krow;
            v2f b;
            b.x = W1[(size_t)kk * H1N + n0 + ml];
            b.y = W1[(size_t)(kk + 1) * H1N + n0 + ml];
            acc = __builtin_amdgcn_wmma_f32_16x16x4_f32(false, a, false, b,
                                                        (short)0, acc, false, false);
        }
        ASYNC_WAIT();          // next chunk resident in LDS (this wave's part)
        __syncthreads();       // all waves done staging + done reading cur
    }
    {   // bias + LeakyReLU(0.2) -> LDS h1
        float bv = b1[n0 + ml];
        #pragma unroll
        for (int r = 0; r < 8; ++r) {
            float v = acc[r] + bv;
            v = v > 0.0f ? v : 0.2f * v;
            h1s[(r + rbase) * H1_PITCH + n0 + ml] = v;
        }
    }
    __syncthreads();

    // ---------------- GEMM2: h2 = h1 @ W2 + b2, tile 16x32 (waves 0,1) --------
    if (wave < 2) {
        const int n2 = wave * 16;
        v8f c2 = {};
        #pragma unroll
        for (int k4 = 0; k4 < H1N; k4 += 4) {
            float2 af = *(const float2*)(&h1s[ml * H1_PITCH + k4 + krow]);
            v2f a; a.x = af.x; a.y = af.y;
            int kk = k4 + krow;
            v2f b;
            b.x = W2[kk * H2N + n2 + ml];
            b.y = W2[(kk + 1) * H2N + n2 + ml];
            c2 = __builtin_amdgcn_wmma_f32_16x16x4_f32(false, a, false, b,
                                                       (short)0, c2, false, false);
        }
        float bv = b2[n2 + ml];
        #pragma unroll
        for (int r = 0; r < 8; ++r)
            h2s[(r + rbase) * H2_PITCH + n2 + ml] = c2[r] + bv;
    }
    __syncthreads();

    // ---------------- Head GEMMs: [16,32] @ [32, 720] in 45 N-tiles -----------
    for (int t = wave; t < 45; t += 4) {       // wave-uniform
        const float* Wt; const float* bt; int nh; int obase; int ncols;
        if (t < 8)       { Wt = Wp;  bt = bp;  nh = t * 16;        obase = OFF_P;  ncols = 128; }
        else if (t < 40) { Wt = Wph; bt = bph; nh = (t - 8) * 16;  obase = OFF_PH; ncols = 512; }
        else if (t == 40){ Wt = Wz;  bt = bz;  nh = 0;             obase = OFF_Z;  ncols = 16;  }
        else             { Wt = Wb;  bt = bb;  nh = (t - 41) * 16; obase = OFF_B;  ncols = 64;  }
        v8f ch = {};
        #pragma unroll
        for (int k4 = 0; k4 < H2N; k4 += 4) {
            float2 af = *(const float2*)(&h2s[ml * H2_PITCH + k4 + krow]);
            v2f a; a.x = af.x; a.y = af.y;
            int kk = k4 + krow;
            v2f b;
            b.x = Wt[kk * ncols + nh + ml];
            b.y = Wt[(kk + 1) * ncols + nh + ml];
            ch = __builtin_amdgcn_wmma_f32_16x16x4_f32(false, a, false, b,
                                                       (short)0, ch, false, false);
        }
        float bv = bt[nh + ml];
        #pragma unroll
        for (int r = 0; r < 8; ++r)
            hh[(r + rbase) * HH_PITCH + obase + nh + ml] = ch[r] + bv;
    }
    __syncthreads();

    // ---------------- Epilogue ------------------------------------------------
    // wp = sigmoid(|p|); per-row sum (8 threads x 16 entries per row)
    {
        int row = tid >> 3, seg = tid & 7;
        float s = 0.0f;
        #pragma unroll
        for (int j = 0; j < 16; ++j) {
            int c = seg * 16 + j;
            float v = sigm(fabsf(hh[row * HH_PITCH + OFF_P + c]));
            hh[row * HH_PITCH + OFF_P + c] = v;   // overwrite preact with sigmoid
            s += v;
        }
        partial[row * 8 + seg] = s;
    }
    __syncthreads();
    if (tid < ROWS) {
        float s = 0.0f;
        #pragma unroll
        for (int i = 0; i < 8; ++i) s = s + partial[tid * 8 + i];
        rowsum[tid] = s;
    }
    __syncthreads();

    // W output: complex [16 rows x 512]; 8192 complex per block
    for (int idx = tid; idx < ROWS * 512; idx += 128) {
        int row = idx >> 9;
        int r   = idx & 511;                  // ((m*8+k)*4 + t)
        float wp_s  = hh[row * HH_PITCH + OFF_P + (r >> 2)];
        float wpn   = wp_s / (rowsum[row] + 1e-8f) * 21.0f;   // PMAX*0.7
        float wmag  = sqrtf(wpn * 0.25f);                     // /Nt
        float ph    = tanhf(hh[row * HH_PITCH + OFF_PH + r]) * 3.14159265358979323846f;
        float sn, cs;
        __sincosf(ph, &sn, &cs);
        size_t o = W_OFF + ((size_t)(row0 + row) * 512 + r) * 2;
        out[o]     = wmag * cs;
        out[o + 1] = wmag * sn;
    }

    // b output: [16 rows x 64]
    for (int idx = tid; idx < ROWS * 64; idx += 128) {
        int row = idx >> 6, n = idx & 63;
        out[B_OFF + (size_t)(row0 + row) * 64 + n] = sigm(hh[row * HH_PITCH + OFF_B + n]);
    }

    // Z output: from batch row 0 only (block 0)
    if (blockIdx.x == 0 && tid == 0) {
        float zs = 0.0f, zsig[16];
        #pragma unroll
        for (int m = 0; m < 16; ++m) {
            float v = sigm(fabsf(hh[0 * HH_PITCH + OFF_Z + m]));
            zsig[m] = v; zs += v;
        }
        float scale = 9.0f / (zs + 1e-8f);    // PMAX*0.3 / sum
        for (int m = 0; m < 16; ++m) {
            float zd = zsig[m] * scale * 0.25f;   // / Nt
            #pragma unroll
            for (int i = 0; i < 4; ++i)
                #pragma unroll
                for (int j = 0; j < 4; ++j) {
                    size_t o = Z_OFF + ((size_t)(m * 16 + i * 4 + j)) * 2;
                    out[o]     = (i == j) ? zd : 0.0f;
                    out[o + 1] = 0.0f;
                }
        }
    }
}

extern "C" void kernel_launch(void* const* d_in, const int* in_sizes, int n_in,
                              void* d_out, int out_size, void* d_ws, size_t ws_size,
                              hipStream_t stream) {
    const float* x   = (const float*)d_in[0];
    const float* W1  = (const float*)d_in[1];
    const float* b1  = (const float*)d_in[2];
    const float* W2  = (const float*)d_in[3];
    const float* b2  = (const float*)d_in[4];
    const float* Wp  = (const float*)d_in[5];
    const float* bp  = (const float*)d_in[6];
    const float* Wph = (const float*)d_in[7];
    const float* bph = (const float*)d_in[8];
    const float* Wz  = (const float*)d_in[9];
    const float* bz  = (const float*)d_in[10];
    const float* Wb  = (const float*)d_in[11];
    const float* bb  = (const float*)d_in[12];
    float* out = (float*)d_out;

    dim3 grid(BATCH / ROWS);   // 512
    dim3 block(128);           // 4 wave32
    isacnet_fused<<<grid, block, 0, stream>>>(x, W1, b1, W2, b2, Wp, bp,
                                              Wph, bph, Wz, bz, Wb, bb, out);
}